// GStar4L4WNet_76871324664554
// MI455X (gfx1250) — compile-verified
//
#include <hip/hip_runtime.h>
#include <math.h>

typedef __attribute__((ext_vector_type(2))) float v2f;
typedef __attribute__((ext_vector_type(8))) float v8f;

#define NBLK 1024
#define EPS_BN 1e-5f

// D = A(16xK) * B(KxN) + C accumulated via V_WMMA_F32_16X16X4_F32 chains.
// A rows = batch rows (M), B = W^T zero-padded, C/D = h tile.
template<int DIN, int DOUT, bool FIRST, bool LAST>
__global__ __launch_bounds__(256) void layer_kernel(
    const float* __restrict__ in,     // [B, DIN] pre-norm activations (or x if FIRST)
    const float* __restrict__ W,      // [DOUT, DIN]
    const float* __restrict__ bias,   // [DOUT]
    const float* __restrict__ stat,   // [2*DIN]: scale, shift (unused if FIRST)
    float* __restrict__ out,          // [B, DOUT] pre-norm h (or [B] if LAST)
    float* __restrict__ partials,     // [NBLK, 2*DOUT] (unused if LAST)
    int batch)
{
  constexpr int KS = (DIN + 3) / 4;    // K slices of 4
  constexpr int NC = (DOUT + 15) / 16; // N chunks of 16
  const int tid = threadIdx.x;
  const int lane = tid & 31;
  const int wv  = tid >> 5;
  const int hl  = lane >> 4;   // half: 0 or 1
  const int lm  = lane & 15;

  // ---- B fragments (W^T tile, zero padded), hoisted out of batch loop ----
  v2f bfrag[NC][KS];
  #pragma unroll
  for (int nc = 0; nc < NC; ++nc) {
    const int n = nc * 16 + lm;
    #pragma unroll
    for (int ks = 0; ks < KS; ++ks)
      #pragma unroll
      for (int r = 0; r < 2; ++r) {
        const int k = ks * 4 + hl * 2 + r;
        bfrag[nc][ks][r] = (n < DOUT && k < DIN) ? W[n * DIN + k] : 0.0f;
      }
  }
  float bcol[NC];
  #pragma unroll
  for (int nc = 0; nc < NC; ++nc) {
    const int n = nc * 16 + lm;
    bcol[nc] = (n < DOUT) ? bias[n] : 0.0f;
  }
  // ---- per-lane BN scale/shift for the input columns this lane touches ----
  float sc[KS][2], sh[KS][2];
  if (!FIRST) {
    #pragma unroll
    for (int ks = 0; ks < KS; ++ks)
      #pragma unroll
      for (int r = 0; r < 2; ++r) {
        const int k = ks * 4 + hl * 2 + r;
        sc[ks][r] = (k < DIN) ? stat[k] : 0.0f;
        sh[ks][r] = (k < DIN) ? stat[DIN + k] : 0.0f;
      }
  }

  float sacc[NC], qacc[NC];
  #pragma unroll
  for (int nc = 0; nc < NC; ++nc) { sacc[nc] = 0.0f; qacc[nc] = 0.0f; }

  // ---- grid-stride over 128-row block tiles; each wave owns a 16-row tile ----
  for (int base = blockIdx.x * 128 + wv * 16; base < batch; base += NBLK * 128) {
    const long long rowA = (long long)base + lm;   // A-matrix row for this lane
    v8f c[NC];
    #pragma unroll
    for (int nc = 0; nc < NC; ++nc)
      #pragma unroll
      for (int j = 0; j < 8; ++j) c[nc][j] = 0.0f;

    #pragma unroll
    for (int ks = 0; ks < KS; ++ks) {
      v2f a;
      #pragma unroll
      for (int r = 0; r < 2; ++r) {
        const int k = ks * 4 + hl * 2 + r;
        float v = (k < DIN) ? in[rowA * DIN + k] : 0.0f;
        if (!FIRST) v = (k < DIN) ? tanhf(v * sc[ks][r] + sh[ks][r]) : 0.0f;
        a[r] = v;
      }
      #pragma unroll
      for (int nc = 0; nc < NC; ++nc)
        c[nc] = __builtin_amdgcn_wmma_f32_16x16x4_f32(
            false, a, false, bfrag[nc][ks], (short)0, c[nc], false, false);
    }

    // ---- epilogue: bias, store pre-norm h, accumulate batch stats ----
    #pragma unroll
    for (int nc = 0; nc < NC; ++nc) {
      const int n = nc * 16 + lm;
      float sv = 0.0f, qv = 0.0f;
      #pragma unroll
      for (int r = 0; r < 8; ++r) {
        const float v = c[nc][r] + bcol[nc];
        const long long row = (long long)base + hl * 8 + r;  // D: VGPR r -> M = r + 8*half
        if (LAST) {
          if (n == 0) out[row] = 1.0f / (1.0f + expf(-v));
        } else {
          if (n < DOUT) {
            out[row * DOUT + n] = v;
            sv += v; qv += v * v;
          }
        }
      }
      sacc[nc] += sv; qacc[nc] += qv;
    }
  }

  if (!LAST) {
    // deterministic reduction: halves via shfl_xor(16), waves via LDS in fixed order
    __shared__ float reds[8 * DOUT];
    __shared__ float redq[8 * DOUT];
    #pragma unroll
    for (int nc = 0; nc < NC; ++nc) {
      const float s = sacc[nc] + __shfl_xor(sacc[nc], 16);
      const float q = qacc[nc] + __shfl_xor(qacc[nc], 16);
      const int n = nc * 16 + lm;
      if (hl == 0 && n < DOUT) { reds[wv * DOUT + n] = s; redq[wv * DOUT + n] = q; }
    }
    __syncthreads();
    if (tid < DOUT) {
      float s = 0.0f, q = 0.0f;
      #pragma unroll
      for (int w = 0; w < 8; ++w) { s += reds[w * DOUT + tid]; q += redq[w * DOUT + tid]; }
      partials[blockIdx.x * (2 * DOUT) + tid] = s;
      partials[blockIdx.x * (2 * DOUT) + DOUT + tid] = q;
    }
  }
}

// Fold block partials (fixed order) -> BN scale/shift for the next layer.
template<int D>
__global__ void stats_kernel(const float* __restrict__ partials,
                             const float* __restrict__ g,
                             const float* __restrict__ bt,
                             float* __restrict__ stat, float inv_batch)
{
  const int j = threadIdx.x;
  if (j >= D) return;
  float s = 0.0f, q = 0.0f;
  for (int b = 0; b < NBLK; ++b) {
    s += partials[b * (2 * D) + j];
    q += partials[b * (2 * D) + D + j];
  }
  const float m   = s * inv_batch;
  const float var = fmaxf(q * inv_batch - m * m, 0.0f);  // biased batch var
  const float scl = g[j] * rsqrtf(var + EPS_BN);
  stat[j]     = scl;
  stat[D + j] = bt[j] - m * scl;
}

extern "C" void kernel_launch(void* const* d_in, const int* in_sizes, int n_in,
                              void* d_out, int out_size, void* d_ws, size_t ws_size,
                              hipStream_t stream) {
  const int B = in_sizes[0] / 64;
  const float invB = 1.0f / (float)B;

  const float* x = (const float*)d_in[0];
  const float *Wp[11], *bp[11], *gp[10], *btp[10];
  for (int i = 0; i < 10; ++i) {
    Wp[i]  = (const float*)d_in[1 + 4 * i];
    bp[i]  = (const float*)d_in[2 + 4 * i];
    gp[i]  = (const float*)d_in[3 + 4 * i];
    btp[i] = (const float*)d_in[4 + 4 * i];
  }
  Wp[10] = (const float*)d_in[41];
  bp[10] = (const float*)d_in[42];

  float* ws   = (float*)d_ws;
  float* hA   = ws;                          // holds even-layer outputs (<= B*10)
  float* hB   = ws + (size_t)B * 10;         // holds odd-layer outputs  (<= B*20)
  float* part = hB + (size_t)B * 20;         // NBLK * 40
  float* st   = part + (size_t)NBLK * 40;    // 10 layers * 40
  float* outp = (float*)d_out;

  dim3 grid(NBLK), blk(256);
  layer_kernel<64, 4, true,  false><<<grid, blk, 0, stream>>>(x,  Wp[0],  bp[0],  nullptr,   hA, part, B);
  stats_kernel<4 ><<<1, 32, 0, stream>>>(part, gp[0], btp[0], st + 0 * 40, invB);
  layer_kernel< 4, 20, false, false><<<grid, blk, 0, stream>>>(hA, Wp[1],  bp[1],  st + 0*40, hB, part, B);
  stats_kernel<20><<<1, 32, 0, stream>>>(part, gp[1], btp[1], st + 1 * 40, invB);
  layer_kernel<20, 10, false, false><<<grid, blk, 0, stream>>>(hB, Wp[2],  bp[2],  st + 1*40, hA, part, B);
  stats_kernel<10><<<1, 32, 0, stream>>>(part, gp[2], btp[2], st + 2 * 40, invB);
  layer_kernel<10, 10, false, false><<<grid, blk, 0, stream>>>(hA, Wp[3],  bp[3],  st + 2*40, hB, part, B);
  stats_kernel<10><<<1, 32, 0, stream>>>(part, gp[3], btp[3], st + 3 * 40, invB);
  layer_kernel<10, 10, false, false><<<grid, blk, 0, stream>>>(hB, Wp[4],  bp[4],  st + 3*40, hA, part, B);
  stats_kernel<10><<<1, 32, 0, stream>>>(part, gp[4], btp[4], st + 4 * 40, invB);
  layer_kernel<10, 10, false, false><<<grid, blk, 0, stream>>>(hA, Wp[5],  bp[5],  st + 4*40, hB, part, B);
  stats_kernel<10><<<1, 32, 0, stream>>>(part, gp[5], btp[5], st + 5 * 40, invB);
  layer_kernel<10, 10, false, false><<<grid, blk, 0, stream>>>(hB, Wp[6],  bp[6],  st + 5*40, hA, part, B);
  stats_kernel<10><<<1, 32, 0, stream>>>(part, gp[6], btp[6], st + 6 * 40, invB);
  layer_kernel<10,  5, false, false><<<grid, blk, 0, stream>>>(hA, Wp[7],  bp[7],  st + 6*40, hB, part, B);
  stats_kernel< 5><<<1, 32, 0, stream>>>(part, gp[7], btp[7], st + 7 * 40, invB);
  layer_kernel< 5,  5, false, false><<<grid, blk, 0, stream>>>(hB, Wp[8],  bp[8],  st + 7*40, hA, part, B);
  stats_kernel< 5><<<1, 32, 0, stream>>>(part, gp[8], btp[8], st + 8 * 40, invB);
  layer_kernel< 5,  5, false, false><<<grid, blk, 0, stream>>>(hA, Wp[9],  bp[9],  st + 8*40, hB, part, B);
  stats_kernel< 5><<<1, 32, 0, stream>>>(part, gp[9], btp[9], st + 9 * 40, invB);
  layer_kernel< 5,  1, false, true ><<<grid, blk, 0, stream>>>(hB, Wp[10], bp[10], st + 9*40, outp, nullptr, B);
}